// TotalLoss_2903397892942
// MI455X (gfx1250) — compile-verified
//
#include <hip/hip_runtime.h>
#include <hip/hip_bf16.h>
#include <math.h>

#define BB 8
#define N_EVT 100000
#define N_PARAMS 4000000
#define F32_EPS 1.1920928955078125e-07f

// workspace float layout:
//   per level l: 32 configs (b,pol,dir) * [num(HW) | den(HW)]
//   level float offsets: 0, 65536, 327680, 1376256 ; end = 5570560
//   then 32 floats of t_ref: [(b*2+pol)*2 + {0:min,1:max}]
#define ND_FLOATS 5570560
#define TREF_OFF  5570560

typedef __attribute__((ext_vector_type(2))) float v2f;
typedef __attribute__((ext_vector_type(8))) float v8f;

__device__ __forceinline__ void atomAddF(float* p, float v) {
  __hip_atomic_fetch_add(p, v, __ATOMIC_RELAXED, __HIP_MEMORY_SCOPE_AGENT);
}

// Wave32 sum reduction through V_WMMA_F32_16X16X4_F32.
// A = ones(16x4)  ->  D[m][n] = sum_k B[k][n].
// Row 0 of D sits in C/D VGPR0 lanes 0..15 (M=0) and lanes 16..31 (M=8, same
// column sums). Summing a 16-lane half of d0 yields the total of all B
// entries regardless of how B's 64 slots map to lanes (we park the per-lane
// partial in b[0], zero in b[1]). EXEC must be all ones at the call site.
__device__ __forceinline__ float wave_reduce_sum(float v) {
#if defined(__HIP_DEVICE_COMPILE__)
  v2f a; a[0] = 1.0f; a[1] = 1.0f;
  v2f b; b[0] = v;    b[1] = 0.0f;
  v8f c = {0.f, 0.f, 0.f, 0.f, 0.f, 0.f, 0.f, 0.f};
  c = __builtin_amdgcn_wmma_f32_16x16x4_f32(false, a, false, b, (short)0, c,
                                            false, false);
  float s = c[0];
  s += __shfl_xor(s, 1);
  s += __shfl_xor(s, 2);
  s += __shfl_xor(s, 4);
  s += __shfl_xor(s, 8);
  return s;   // every lane holds the wave total
#else
  return v;   // host pass: never executed
#endif
}

__device__ __forceinline__ void block_reduce_add(float v, float* out,
                                                 float scale) {
  v = wave_reduce_sum(v);
  __shared__ float lds[32];
  int lane = threadIdx.x & 31, wv = threadIdx.x >> 5;
  if (lane == 0) lds[wv] = v;
  __syncthreads();
  if (threadIdx.x == 0) {
    float s = 0.f;
    int nw = (blockDim.x + 31) >> 5;
    for (int w = 0; w < nw; ++w) s += lds[w];
    atomAddF(out, s * scale);
  }
}

// ---------------- init: zero num/den, seed t_ref sentinels, zero out -------
__global__ void k_init(float* __restrict__ ws, float* __restrict__ out) {
  long i = (long)blockIdx.x * blockDim.x + threadIdx.x;
  long stride = (long)gridDim.x * blockDim.x;
  for (long j = i; j < (long)ND_FLOATS; j += stride) ws[j] = 0.0f;
  if (i < 32) {
    int* tr = (int*)(ws + TREF_OFF);
    tr[i] = (i & 1) ? (int)0xFF800000u /* -inf (max slot) */
                    : (int)0x7F800000u /* +inf (min slot) */;
  }
  if (i == 0) out[0] = 0.0f;
}

// ---------------- t_ref: min/max of t per (batch, polarity) ----------------
// ts sorted ascending, so t_ref(fwd)=max over mask, t_ref(bwd)=min over mask.
// t >= 0, so int-bit-pattern atomicMin/Max == float min/max (and the -inf /
// +inf sentinels order correctly as signed ints).
__global__ void k_tref(const float* __restrict__ ts, const int* __restrict__ ps,
                       float* __restrict__ ws) {
  int b = blockIdx.y;
  float minP = INFINITY, maxP = -INFINITY, minN = INFINITY, maxN = -INFINITY;
  for (int e = blockIdx.x * blockDim.x + threadIdx.x; e < N_EVT;
       e += blockDim.x * gridDim.x) {
    float t = ts[b * N_EVT + e];
    if (ps[b * N_EVT + e] == 1) {
      minP = fminf(minP, t); maxP = fmaxf(maxP, t);
    } else {
      minN = fminf(minN, t); maxN = fmaxf(maxN, t);
    }
  }
  for (int m = 16; m; m >>= 1) {
    minP = fminf(minP, __shfl_xor(minP, m));
    maxP = fmaxf(maxP, __shfl_xor(maxP, m));
    minN = fminf(minN, __shfl_xor(minN, m));
    maxN = fmaxf(maxN, __shfl_xor(maxN, m));
  }
  __shared__ float s[4][32];
  int lane = threadIdx.x & 31, wv = threadIdx.x >> 5;
  if (lane == 0) { s[0][wv]=minP; s[1][wv]=maxP; s[2][wv]=minN; s[3][wv]=maxN; }
  __syncthreads();
  if (threadIdx.x == 0) {
    float a = INFINITY, mx = -INFINITY, c = INFINITY, dx = -INFINITY;
    int nw = (blockDim.x + 31) >> 5;
    for (int w = 0; w < nw; ++w) {
      a  = fminf(a,  s[0][w]); mx = fmaxf(mx, s[1][w]);
      c  = fminf(c,  s[2][w]); dx = fmaxf(dx, s[3][w]);
    }
    int* tr = (int*)(ws + TREF_OFF);
    atomicMin(&tr[(b * 2 + 0) * 2 + 0], __float_as_int(a));
    atomicMax(&tr[(b * 2 + 0) * 2 + 1], __float_as_int(mx));
    atomicMin(&tr[(b * 2 + 1) * 2 + 0], __float_as_int(c));
    atomicMax(&tr[(b * 2 + 1) * 2 + 1], __float_as_int(dx));
  }
}

// ---------------- event scatter: bilinear atomic accumulation --------------
// Events of the other polarity have mf==0 and contribute exactly 0 -> each
// event only touches its own polarity's 8 (level,dir) configs.
__global__ void k_scatter(const float* __restrict__ f0,
                          const float* __restrict__ f1,
                          const float* __restrict__ f2,
                          const float* __restrict__ f3,
                          const int* __restrict__ xs, const int* __restrict__ ys,
                          const float* __restrict__ ts, const int* __restrict__ ps,
                          float* __restrict__ ws) {
  int e = blockIdx.x * blockDim.x + threadIdx.x;
  int b = blockIdx.y;
  if (e >= N_EVT) return;
  int gi = b * N_EVT + e;
  int x = xs[gi], y = ys[gi];
  float t = ts[gi];
  int pol = (ps[gi] == 1) ? 0 : 1;
  const float* tr = ws + TREF_OFF + (size_t)(b * 2 + pol) * 2;
  float tmin = tr[0], tmax = tr[1];
  const float* flows[4] = {f0, f1, f2, f3};
  const int loff[4] = {0, 65536, 327680, 1376256};
#pragma unroll
  for (int l = 0; l < 4; ++l) {
    int S = 32 << l;          // level is SxS
    int sh = 3 - l;           // x/scale with trunc == x >> sh for x>=0
    int xi = x >> sh, yi = y >> sh;
    const float* fb = flows[l] + (size_t)(b * 2) * S * S;
    float fx = fb[yi * S + xi];
    float fy = fb[S * S + yi * S + xi];
    float Wm1 = (float)(S - 1);
    int cfgbase = loff[l] + ((b * 2 + pol) * 2) * (2 * S * S);
#pragma unroll
    for (int d = 0; d < 2; ++d) {
      float t_ = (d == 0) ? (tmax - t + F32_EPS) : (tmin - t - F32_EPS);
      float x_ = fminf(fmaxf((float)xi + t_ * fx, 0.0f), Wm1);
      float y_ = fminf(fmaxf((float)yi + t_ * fy, 0.0f), Wm1);
      float x0 = floorf(x_), y0 = floorf(y_);
      float x1 = x0 + 1.0f, y1 = y0 + 1.0f;
      float x0w = x_ - x0, x1w = x1 - x_;
      float y0w = y_ - y0, y1w = y1 - y_;
      float Ta = x0w * y0w * t, Tb = x1w * y0w * t;
      float Tc = x0w * y1w * t, Td = x1w * y1w * t;
      int x1i = (int)fminf(x1, Wm1), y1i = (int)fminf(y1, Wm1);
      int x0i = (int)x0, y0i = (int)y0;
      int Ia = x1i + y1i * S, Ib = x0i + y1i * S;
      int Ic = x1i + y0i * S, Id = x0i + y0i * S;
      float* base = ws + cfgbase + d * (2 * S * S);
      atomAddF(base + Ia, Ta); atomAddF(base + Ib, Tb);
      atomAddF(base + Ic, Tc); atomAddF(base + Id, Td);
      float* den = base + S * S;
      atomAddF(den + Ia, 1.0f); atomAddF(den + Ib, 1.0f);
      atomAddF(den + Ic, 1.0f); atomAddF(den + Id, 1.0f);
    }
  }
}

// ---------------- per-pixel loss reduce over all configs/levels ------------
#define FLOWLOSS_TOTAL 2785280
__global__ void k_flowloss(const float* __restrict__ ws, float* __restrict__ out) {
  int i = blockIdx.x * blockDim.x + threadIdx.x;
  float v = 0.0f;
  if (i < FLOWLOSS_TOTAL) {
    int hw, loff, rem;
    if (i < 32768)        { hw = 1024;  loff = 0;       rem = i; }
    else if (i < 163840)  { hw = 4096;  loff = 65536;   rem = i - 32768; }
    else if (i < 688128)  { hw = 16384; loff = 327680;  rem = i - 163840; }
    else                  { hw = 65536; loff = 1376256; rem = i - 688128; }
    int cfg = rem / hw;
    int pix = rem - cfg * hw;
    const float* bp = ws + loff + (size_t)cfg * (2 * hw);
    float num = bp[pix];
    float den = bp[hw + pix];
    float r = num / (den + F32_EPS);
    v = sqrtf(r * r + 1e-6f);
  }
  block_reduce_add(v, out, 1.0f);   // EXEC reconverged: all lanes present
}

// ---------------- weight decay: sum(p^2)/2 * 1e-4 --------------------------
__global__ void k_wd(const float* __restrict__ p, float* __restrict__ out) {
  float s = 0.f;
  int stride = blockDim.x * gridDim.x;
  for (int i = blockIdx.x * blockDim.x + threadIdx.x; i < N_PARAMS; i += stride) {
    float x = p[i];
    s += x * x;
  }
  block_reduce_add(s, out, 0.00005f);
}

// ---------------- smoothness (Charbonnier) * 25 ----------------------------
#define SMOOTH_TOTAL 1392640
__global__ void k_smooth(const float* __restrict__ f0, const float* __restrict__ f1,
                         const float* __restrict__ f2, const float* __restrict__ f3,
                         float* __restrict__ out) {
  int i = blockIdx.x * blockDim.x + threadIdx.x;
  float acc = 0.f;
  if (i < SMOOTH_TOTAL) {
    const float* fp; int S, rem;
    if (i < 16384)        { fp = f0; S = 32;  rem = i; }
    else if (i < 81920)   { fp = f1; S = 64;  rem = i - 16384; }
    else if (i < 344064)  { fp = f2; S = 128; rem = i - 81920; }
    else                  { fp = f3; S = 256; rem = i - 344064; }
    int x = rem % S;
    int yz = rem / S;
    int y = yz % S;
    int bc = yz / S;                       // 0..15 = (b,c)
    const float* base = fp + (size_t)bc * S * S;
    int Sm1 = S - 1;
    // final = 200 * (0.5/4) * (lr+ud+d1+d2)/4  ->  6.25 * each mean
    float c_edge = 6.25f / (float)(16 * S * Sm1);
    float c_diag = 6.25f / (float)(16 * Sm1 * Sm1);
    float f00 = base[y * S + x];
    bool hx = x < Sm1, hy = y < Sm1;
    float f01 = 0.f, f10 = 0.f, f11 = 0.f;
    if (hx) f01 = base[y * S + x + 1];
    if (hy) f10 = base[(y + 1) * S + x];
    if (hx && hy) f11 = base[(y + 1) * S + x + 1];
    if (hx) { float d = f01 - f00; acc += powf(d * d + 1e-6f, 0.45f) * c_edge; }
    if (hy) { float d = f10 - f00; acc += powf(d * d + 1e-6f, 0.45f) * c_edge; }
    if (hx && hy) {
      float d1 = f11 - f00; acc += powf(d1 * d1 + 1e-6f, 0.45f) * c_diag;
      float d2 = f01 - f10; acc += powf(d2 * d2 + 1e-6f, 0.45f) * c_diag;
    }
  }
  block_reduce_add(acc, out, 1.0f);
}

extern "C" void kernel_launch(void* const* d_in, const int* in_sizes, int n_in,
                              void* d_out, int out_size, void* d_ws, size_t ws_size,
                              hipStream_t stream) {
  (void)in_sizes; (void)n_in; (void)out_size; (void)ws_size;
  const float* f0 = (const float*)d_in[0];
  const float* f1 = (const float*)d_in[1];
  const float* f2 = (const float*)d_in[2];
  const float* f3 = (const float*)d_in[3];
  const int*   xs = (const int*)d_in[4];
  const int*   ys = (const int*)d_in[5];
  const float* ts = (const float*)d_in[6];
  const int*   ps = (const int*)d_in[7];
  // d_in[8], d_in[9] (frame, frame_) are unused by the reference.
  const float* params = (const float*)d_in[10];
  float* ws = (float*)d_ws;
  float* out = (float*)d_out;

  k_init<<<2048, 256, 0, stream>>>(ws, out);
  k_tref<<<dim3(64, BB), 256, 0, stream>>>(ts, ps, ws);
  k_scatter<<<dim3((N_EVT + 255) / 256, BB), 256, 0, stream>>>(
      f0, f1, f2, f3, xs, ys, ts, ps, ws);
  k_flowloss<<<(FLOWLOSS_TOTAL + 255) / 256, 256, 0, stream>>>(ws, out);
  k_wd<<<1024, 256, 0, stream>>>(params, out);
  k_smooth<<<(SMOOTH_TOTAL + 255) / 256, 256, 0, stream>>>(f0, f1, f2, f3, out);
}